// PointConvDownSampling_74552042324074
// MI455X (gfx1250) — compile-verified
//
#include <hip/hip_runtime.h>
#include <hip/hip_bf16.h>

typedef __attribute__((ext_vector_type(2))) float v2f;
typedef __attribute__((ext_vector_type(8))) float v8f;

#define NB     2
#define NPT    16384
#define NS     4096
#define NC     64
#define NOUT   128
#define NK     16
#define CH     512
#define NCHUNK 8
#define CHPTS  (NPT / NCHUNK)     // 2048 points scanned per partial block
#define AGROW  1076               // 1072 flat agg row + 4 pad floats (conflict-free stride)

// ---------------------------------------------------------------------------
// Kernel 1a: partial brute-force KNN. Each block owns 256 query points and
// scans ONE chunk of 2048 candidate points (streamed through LDS, broadcast
// reads). Emits a per-chunk top-16 (dist,idx) list. 256 blocks total -> 8x
// the parallelism of a monolithic scan, which dominates end-to-end time.
// ---------------------------------------------------------------------------
__global__ __launch_bounds__(256) void knn_partial_kernel(
    const float* __restrict__ xyz, const float* __restrict__ sxyz,
    float* __restrict__ pd, int* __restrict__ pi) {
    const int tid = threadIdx.x;
    const int blkPerB = NS / 256;                     // 16
    const int chunk = blockIdx.x % NCHUNK;
    const int t     = blockIdx.x / NCHUNK;
    const int b     = t / blkPerB;
    const int s     = (t % blkPerB) * 256 + tid;

    const float sx = sxyz[(b * 3 + 0) * NS + s];
    const float sy = sxyz[(b * 3 + 1) * NS + s];
    const float sz = sxyz[(b * 3 + 2) * NS + s];
    const float s2 = sx * sx + sy * sy + sz * sz;

    __shared__ float xl[CH], yl[CH], zl[CH], x2l[CH];

    float bd[NK];
    int   bi[NK];
#pragma unroll
    for (int i = 0; i < NK; ++i) { bd[i] = 3.4e38f; bi[i] = 0; }

    const float* xb = xyz + (size_t)b * 3 * NPT;
    const int n0 = chunk * CHPTS;

    for (int c0 = n0; c0 < n0 + CHPTS; c0 += CH) {
        __syncthreads();
        for (int j = tid; j < CH; j += 256) {
            float xv = xb[c0 + j];
            float yv = xb[NPT + c0 + j];
            float zv = xb[2 * NPT + c0 + j];
            xl[j] = xv; yl[j] = yv; zl[j] = zv;
            x2l[j] = xv * xv + yv * yv + zv * zv;
        }
        __syncthreads();
#pragma unroll 4
        for (int jj = 0; jj < CH; ++jj) {
            float dot = sx * xl[jj] + sy * yl[jj] + sz * zl[jj];
            float d2  = (s2 + x2l[jj]) - 2.0f * dot;   // reference's expansion form
            if (d2 < bd[NK - 1]) {
                float cd = d2; int ci = c0 + jj;
#pragma unroll
                for (int i = 0; i < NK; ++i) {
                    if (cd < bd[i]) {
                        float td = bd[i]; int ti = bi[i];
                        bd[i] = cd; bi[i] = ci;
                        cd = td; ci = ti;
                    }
                }
            }
        }
    }
    const size_t base = ((size_t)(b * NS + s) * NCHUNK + chunk) * NK;
#pragma unroll
    for (int i = 0; i < NK; ++i) { pd[base + i] = bd[i]; pi[base + i] = bi[i]; }
}

// ---------------------------------------------------------------------------
// Kernel 1b: merge the NCHUNK partial top-16 lists (128 candidates) into the
// exact global top-16 per query. One thread per query.
// ---------------------------------------------------------------------------
__global__ __launch_bounds__(256) void knn_merge_kernel(
    const float* __restrict__ pd, const int* __restrict__ pi,
    int* __restrict__ knn) {
    const int g = blockIdx.x * 256 + threadIdx.x;     // 0 .. NB*NS-1
    const int b = g / NS;
    const int s = g % NS;

    float bd[NK];
    int   bi[NK];
#pragma unroll
    for (int i = 0; i < NK; ++i) { bd[i] = 3.4e38f; bi[i] = 0; }

    const size_t base = (size_t)(b * NS + s) * NCHUNK * NK;
    for (int c = 0; c < NCHUNK * NK; ++c) {
        float d = pd[base + c];
        int   ci = pi[base + c];
        if (d < bd[NK - 1]) {
            float cd = d;
#pragma unroll
            for (int i = 0; i < NK; ++i) {
                if (cd < bd[i]) {
                    float td = bd[i]; int ti = bi[i];
                    bd[i] = cd; bi[i] = ci;
                    cd = td; ci = ti;
                }
            }
        }
    }
#pragma unroll
    for (int i = 0; i < NK; ++i)
        knn[((size_t)(b * NS + s)) * NK + i] = bi[i];
}

// ---------------------------------------------------------------------------
// Kernel 2: weight-net + gather + agg (WMMA) + final 128x1072 GEMM (WMMA).
// One workgroup = one batch element x 16 query points; 8 waves (256 threads).
// (Unchanged from round 1 -- codegen verified: v_wmma_f32_16x16x4_f32.)
// ---------------------------------------------------------------------------
__global__ __launch_bounds__(256) void pointconv_kernel(
    const float* __restrict__ xyz,  const float* __restrict__ feat,
    const float* __restrict__ sxyz,
    const float* __restrict__ w1,   const float* __restrict__ b1,
    const float* __restrict__ w2,   const float* __restrict__ b2,
    const float* __restrict__ lw,   const float* __restrict__ lb,
    const int*   __restrict__ knn,  float* __restrict__ out)
{
    const int tid  = threadIdx.x;
    const int lane = tid & 31;
    const int wave = tid >> 5;
    const int b    = blockIdx.x >> 8;          // 256 tiles per batch
    const int s0   = (blockIdx.x & 255) * 16;

    __shared__ float w1s[24], b1s[8], w2s[128], b2s[16];
    __shared__ float Wl[16][16][16];           // [s][k][wi]
    __shared__ float Fl[16][16][68];           // [s][k][c], col 67 = zero pad
    __shared__ float Ag[16 * AGROW];           // [s][wi*67+c] flat, padded row

    if (tid < 24)  w1s[tid] = w1[tid];
    if (tid < 8)   b1s[tid] = b1[tid];
    if (tid < 128) w2s[tid] = w2[tid];
    if (tid < 16)  b2s[tid] = b2[tid];
    __syncthreads();

    // ---- Phase 1: weight-net + feature gather, one thread per (s,k) ----
    {
        const int s = tid >> 4, k = tid & 15;
        const int idx = knn[((size_t)(b * NS + s0 + s)) * NK + k];
        const float nx = xyz[(size_t)b * 3 * NPT + idx];
        const float ny = xyz[(size_t)b * 3 * NPT + NPT + idx];
        const float nz = xyz[(size_t)b * 3 * NPT + 2 * NPT + idx];
        const float rx = nx - sxyz[(b * 3 + 0) * NS + s0 + s];
        const float ry = ny - sxyz[(b * 3 + 1) * NS + s0 + s];
        const float rz = nz - sxyz[(b * 3 + 2) * NS + s0 + s];

        float h[8];
#pragma unroll
        for (int o = 0; o < 8; ++o) {
            float v = w1s[o * 3 + 0] * rx + w1s[o * 3 + 1] * ry +
                      w1s[o * 3 + 2] * rz + b1s[o];
            h[o] = v >= 0.f ? v : 0.1f * v;
        }
#pragma unroll
        for (int o = 0; o < 16; ++o) {
            float v = b2s[o];
#pragma unroll
            for (int i = 0; i < 8; ++i) v += w2s[o * 8 + i] * h[i];
            v = v >= 0.f ? v : 0.1f * v;
            Wl[s][k][o] = v;
        }
        Fl[s][k][0] = nx; Fl[s][k][1] = ny; Fl[s][k][2] = nz;
        const float* fb = feat + (size_t)b * NC * NPT + idx;
#pragma unroll 8
        for (int cf = 0; cf < NC; ++cf)
            Fl[s][k][3 + cf] = fb[(size_t)cf * NPT];
        Fl[s][k][67] = 0.f;
    }
    __syncthreads();

    // ---- Phase 2: agg[s] = W^T (16x16) x F (16x67) via WMMA f32 16x16x4 ----
    // A frag: lane -> M=lane%16 (wi); VGPR0/1 hold K = {0,1} (lanes<16) / {2,3}
    // B frag: lane -> N=lane%16 (c);  same K-per-VGPR mapping.
    {
        const int r    = lane & 15;
        const int koff = (lane < 16) ? 0 : 2;
#pragma unroll
        for (int si = 0; si < 2; ++si) {
            const int s = wave * 2 + si;
#pragma unroll
            for (int ct = 0; ct < 5; ++ct) {
                const int c  = ct * 16 + r;
                const int cc = c > 67 ? 67 : c;        // pad column reads 0
                v8f acc = {};
#pragma unroll
                for (int kb = 0; kb < 4; ++kb) {
                    const int k0 = kb * 4 + koff;
                    v2f a;  a.x  = Wl[s][k0][r];  a.y  = Wl[s][k0 + 1][r];
                    v2f bb; bb.x = Fl[s][k0][cc]; bb.y = Fl[s][k0 + 1][cc];
                    acc = __builtin_amdgcn_wmma_f32_16x16x4_f32(
                        false, a, false, bb, (short)0, acc, false, false);
                }
                if (c < 67) {
#pragma unroll
                    for (int rr = 0; rr < 8; ++rr) {
                        const int wi = rr + ((lane < 16) ? 0 : 8);  // D: M = rr / rr+8
                        Ag[s * AGROW + wi * 67 + c] = acc[rr];
                    }
                }
            }
        }
    }
    __syncthreads();

    // ---- Phase 3: out(128x16) = lin_w(128x1072) x Ag(1072x16), WMMA K-loop ----
    {
        const int r    = lane & 15;
        const int koff = (lane < 16) ? 0 : 2;
        const float* Lrow = lw + (size_t)(wave * 16 + r) * 1072;  // A: M = lane%16
        const float* Arow = &Ag[r * AGROW];                        // B: N = lane%16 (= s)
        v8f acc = {};
        for (int kb = 0; kb < 268; ++kb) {
            const int k0 = kb * 4 + koff;
            v2f a;  a.x  = Lrow[k0]; a.y  = Lrow[k0 + 1];
            v2f bb; bb.x = Arow[k0]; bb.y = Arow[k0 + 1];
            acc = __builtin_amdgcn_wmma_f32_16x16x4_f32(
                false, a, false, bb, (short)0, acc, false, false);
        }
        const int su = s0 + r;
#pragma unroll
        for (int rr = 0; rr < 8; ++rr) {
            const int o = wave * 16 + rr + ((lane < 16) ? 0 : 8);
            float v = acc[rr] + lb[o];
            v = v >= 0.f ? v : 0.1f * v;
            out[(size_t)b * NOUT * NS + (size_t)o * NS + su] = v;
        }
    }
}

// ---------------------------------------------------------------------------
extern "C" void kernel_launch(void* const* d_in, const int* in_sizes, int n_in,
                              void* d_out, int out_size, void* d_ws, size_t ws_size,
                              hipStream_t stream) {
    const float* xyz  = (const float*)d_in[0];
    const float* feat = (const float*)d_in[1];
    const float* sxyz = (const float*)d_in[2];
    const float* w1   = (const float*)d_in[3];
    const float* b1   = (const float*)d_in[4];
    const float* w2   = (const float*)d_in[5];
    const float* b2   = (const float*)d_in[6];
    const float* lw   = (const float*)d_in[7];
    const float* lb   = (const float*)d_in[8];
    float* out = (float*)d_out;

    // workspace layout: partial dists (4MB) | partial idx (4MB) | final knn (512KB)
    const size_t npart = (size_t)NB * NS * NCHUNK * NK;   // 1,048,576
    float* pd  = (float*)d_ws;
    int*   pi  = (int*)((char*)d_ws + npart * sizeof(float));
    int*   knn = (int*)((char*)d_ws + 2 * npart * sizeof(float));

    knn_partial_kernel<<<NB * (NS / 256) * NCHUNK, 256, 0, stream>>>(xyz, sxyz, pd, pi);
    knn_merge_kernel<<<NB * NS / 256, 256, 0, stream>>>(pd, pi, knn);
    pointconv_kernel<<<NB * (NS / 16), 256, 0, stream>>>(
        xyz, feat, sxyz, w1, b1, w2, b2, lw, lb, knn, out);
}